// Loss_fn_21852793602431
// MI455X (gfx1250) — compile-verified
//
#include <hip/hip_runtime.h>
#include <math.h>

#define EPS_ 1e-7f
#define B_ 64
#define N_ 256
#define M_ 256

typedef __attribute__((ext_vector_type(2))) float v2f;
typedef __attribute__((ext_vector_type(8))) float v8f;

__device__ __forceinline__ float diou_loss_dev(const float* b1, const float* b2) {
    float x1 = b1[0], y1 = b1[1], x2 = b1[2], y2 = b1[3];
    float xg1 = b2[0], yg1 = b2[1], xg2 = b2[2], yg2 = b2[3];
    float xi1 = fmaxf(x1, xg1), yi1 = fmaxf(y1, yg1);
    float xi2 = fminf(x2, xg2), yi2 = fminf(y2, yg2);
    float inter = fmaxf(xi2 - xi1, 0.f) * fmaxf(yi2 - yi1, 0.f);
    float uni = (x2 - x1) * (y2 - y1) + (xg2 - xg1) * (yg2 - yg1) - inter;
    float iou_e = inter / (uni + EPS_);
    float xc1 = fminf(x1, xg1), yc1 = fminf(y1, yg1);
    float xc2 = fmaxf(x2, xg2), yc2 = fmaxf(y2, yg2);
    float dx = xc2 - xc1, dy = yc2 - yc1;
    float diag = dx * dx + dy * dy + EPS_;
    float cdx = (x1 + x2 - xg1 - xg2) * 0.5f;
    float cdy = (y1 + y2 - yg1 - yg2) * 0.5f;
    float dist = cdx * cdx + cdy * cdy;
    return 1.f - iou_e + dist / diag;
}

// ---------------------------------------------------------------- prep
// per (b,n): sigmoid, clamped logs, logit diff (transposed for WMMA A),
// cxcywh->xyxy for both pred and target boxes.
__global__ void prep_kernel(const float* __restrict__ labels_pred,
                            const float* __restrict__ bbox_pred,
                            const float* __restrict__ bbox_target,
                            float* __restrict__ pbxy, float* __restrict__ tbxy,
                            float* __restrict__ dT,
                            float* __restrict__ lnparr, float* __restrict__ ln1arr) {
    int idx = blockIdx.x * blockDim.x + threadIdx.x;
    if (idx >= B_ * N_) return;
    int b = idx / N_, n = idx - b * N_;
    float x = labels_pred[idx];
    float p = 1.f / (1.f + expf(-x));
    float lnp = fmaxf(logf(p), -100.f);
    float ln1 = fmaxf(logf(1.f - p), -100.f);
    lnparr[idx] = lnp;
    ln1arr[idx] = ln1;
    dT[n * B_ + b] = lnp - ln1;
    {   // pred box -> xyxy
        float cx = bbox_pred[idx * 4 + 0], cy = bbox_pred[idx * 4 + 1];
        float w = bbox_pred[idx * 4 + 2], h = bbox_pred[idx * 4 + 3];
        pbxy[idx * 4 + 0] = cx - 0.5f * w; pbxy[idx * 4 + 1] = cy - 0.5f * h;
        pbxy[idx * 4 + 2] = cx + 0.5f * w; pbxy[idx * 4 + 3] = cy + 0.5f * h;
    }
    {   // target box -> xyxy (same flat count since N==M)
        float cx = bbox_target[idx * 4 + 0], cy = bbox_target[idx * 4 + 1];
        float w = bbox_target[idx * 4 + 2], h = bbox_target[idx * 4 + 3];
        tbxy[idx * 4 + 0] = cx - 0.5f * w; tbxy[idx * 4 + 1] = cy - 0.5f * h;
        tbxy[idx * 4 + 2] = cx + 0.5f * w; tbxy[idx * 4 + 3] = cy + 0.5f * h;
    }
}

// s1[n] = sum_b ln1[b,n]
__global__ void s1_kernel(const float* __restrict__ ln1arr, float* __restrict__ s1) {
    int n = blockIdx.x * blockDim.x + threadIdx.x;
    if (n >= N_) return;
    float s = 0.f;
    for (int b = 0; b < B_; ++b) s += ln1arr[b * N_ + n];
    s1[n] = s;
}

// ------------------------------------------------ label-cost GEMM (WMMA)
// L[n,m] = sum_b dT[n,b] * t[b,m]   (N=M=256, K=B=64) via v_wmma_f32_16x16x4_f32
__global__ void label_gemm_kernel(const float* __restrict__ dT,
                                  const float* __restrict__ tmat,
                                  float* __restrict__ Lmat) {
    int tile = blockIdx.x;                 // 256 tiles of 16x16, one wave each
    int n0 = (tile / (M_ / 16)) * 16;
    int m0 = (tile % (M_ / 16)) * 16;
    int lane = threadIdx.x;                // 0..31
    int half = lane >> 4;                  // 0 or 1
    int lr = lane & 15;
    v8f c = {};
    for (int k0 = 0; k0 < B_; k0 += 4) {
        int ka = k0 + 2 * half;            // lanes 16-31 hold K=2,3 of the 4-slice
        v2f a, bm;
        a.x = dT[(n0 + lr) * B_ + ka];
        a.y = dT[(n0 + lr) * B_ + ka + 1];
        bm.x = tmat[(ka) * M_ + m0 + lr];
        bm.y = tmat[(ka + 1) * M_ + m0 + lr];
        c = __builtin_amdgcn_wmma_f32_16x16x4_f32(
                /*neg_a=*/false, a, /*neg_b=*/false, bm,
                /*c_mod=*/(short)0, c, /*reuse_a=*/false, /*reuse_b=*/false);
    }
    for (int v = 0; v < 8; ++v) {
        int row = n0 + v + 8 * half;       // lanes 16-31 hold rows M+8
        Lmat[row * M_ + m0 + lr] = c[v];
    }
}

// ------------------------------------- costbase = l1_cost + label_cost  [N,M]
__global__ void __launch_bounds__(256) costbase_kernel(
        const float* __restrict__ bbox_pred, const float* __restrict__ bbox_target,
        const float* __restrict__ Lmat, const float* __restrict__ s1,
        float* __restrict__ costbase) {
    int tile = blockIdx.x;
    int n0 = (tile / (M_ / 16)) * 16, m0 = (tile % (M_ / 16)) * 16;
    int tn = threadIdx.x >> 4, tm = threadIdx.x & 15;
    int n = n0 + tn, m = m0 + tm;
    float acc = 0.f;
    for (int b = 0; b < B_; ++b) {
        const float* bp = bbox_pred + ((size_t)(b * N_ + n)) * 4;
        const float* bt = bbox_target + ((size_t)(b * M_ + m)) * 4;
        __builtin_prefetch(bp + N_ * 4, 0, 0);   // next batch's row
        acc += fabsf(bp[0] - bt[0]) + fabsf(bp[1] - bt[1]) +
               fabsf(bp[2] - bt[2]) + fabsf(bp[3] - bt[3]);
    }
    float l1 = acc * (1.f / (4.f * B_));
    float label = -(Lmat[n * M_ + m] + s1[n]) * (1.f / (float)B_);
    costbase[n * M_ + m] = l1 + label;
}

// ------------------------------------------------ Hungarian, 1 block per batch
// Kuhn-Munkres with potentials, columns parallel across 256 threads (8 waves).
// cost rows are recomputed on the fly: diou(pred[i0], tgt[j]) + costbase[i0,j].
__global__ void __launch_bounds__(256) hungarian_kernel(
        const float* __restrict__ pbxy, const float* __restrict__ tbxy,
        const float* __restrict__ costbase, int* __restrict__ cols) {
    const int b = blockIdx.x;
    const int tid = threadIdx.x;
    const int j = tid + 1;                 // this thread's column, 1..256

    __shared__ float u[N_ + 1], v[N_ + 1], minv[N_ + 1];
    __shared__ int pcol[N_ + 1], way[N_ + 1], usedf[N_ + 1];
    __shared__ float red_v[N_];
    __shared__ int red_j[N_];
    __shared__ float tbs[N_][4];
    __shared__ float pbrow[4];
    __shared__ int j0_s, i0_s, brk_s;
    __shared__ float delta_s;

    const float FINF = __builtin_inff();

    u[j] = 0.f; v[j] = 0.f; pcol[j] = 0; way[j] = 0;
    if (tid == 0) { u[0] = 0.f; v[0] = 0.f; pcol[0] = 0; way[0] = 0; }
    const float* tb = tbxy + (size_t)b * M_ * 4;
    tbs[tid][0] = tb[tid * 4 + 0]; tbs[tid][1] = tb[tid * 4 + 1];
    tbs[tid][2] = tb[tid * 4 + 2]; tbs[tid][3] = tb[tid * 4 + 3];
    __syncthreads();

    for (int i = 1; i <= N_; ++i) {
        if (tid == 0) { pcol[0] = i; j0_s = 0; usedf[0] = 0; }
        minv[j] = FINF; usedf[j] = 0;
        __syncthreads();
        for (;;) {
            if (tid == 0) {
                usedf[j0_s] = 1;
                i0_s = pcol[j0_s];
                const float* pr = pbxy + ((size_t)(b * N_ + i0_s - 1)) * 4;
                pbrow[0] = pr[0]; pbrow[1] = pr[1]; pbrow[2] = pr[2]; pbrow[3] = pr[3];
            }
            __syncthreads();
            int i0 = i0_s, j0 = j0_s;
            if (!usedf[j]) {
                float cij = diou_loss_dev(pbrow, tbs[j - 1]) +
                            costbase[(i0 - 1) * M_ + (j - 1)];
                float cur = cij - u[i0] - v[j];
                if (cur < minv[j]) { minv[j] = cur; way[j] = j0; }
            }
            red_v[tid] = usedf[j] ? FINF : minv[j];
            red_j[tid] = j;
            __syncthreads();
            for (int s = 128; s > 0; s >>= 1) {
                if (tid < s) {
                    float ov = red_v[tid + s]; int oj = red_j[tid + s];
                    if (ov < red_v[tid] ||
                        (ov == red_v[tid] && oj < red_j[tid])) {
                        red_v[tid] = ov; red_j[tid] = oj;   // first-min, like argmin
                    }
                }
                __syncthreads();
            }
            if (tid == 0) { delta_s = red_v[0]; j0_s = red_j[0]; }
            __syncthreads();
            float delta = delta_s;
            if (usedf[j]) { u[pcol[j]] += delta; v[j] -= delta; }
            else          { minv[j] -= delta; }
            if (tid == 0 && usedf[0]) { u[pcol[0]] += delta; v[0] -= delta; }
            __syncthreads();
            if (tid == 0) brk_s = (pcol[j0_s] == 0) ? 1 : 0;
            __syncthreads();
            if (brk_s) break;
        }
        if (tid == 0) {                    // augment path (serial chase)
            int j0 = j0_s;
            while (j0) { int j1 = way[j0]; pcol[j0] = pcol[j1]; j0 = j1; }
        }
        __syncthreads();
    }
    cols[b * N_ + pcol[j] - 1] = j - 1;    // col_of_row
}

// ------------------------------------------- matched losses, partials per batch
__global__ void __launch_bounds__(256) matched_kernel(
        const float* __restrict__ pbxy, const float* __restrict__ tbxy,
        const float* __restrict__ bbox_pred, const float* __restrict__ bbox_target,
        const float* __restrict__ labels_target,
        const float* __restrict__ lnparr, const float* __restrict__ ln1arr,
        const int* __restrict__ cols, float* __restrict__ partials) {
    int b = blockIdx.x, n = threadIdx.x;
    int col = cols[b * N_ + n];
    float t_m = labels_target[b * M_ + col];
    float w = (t_m == 1.0f) ? 1.f : 0.f;

    const float* b1 = pbxy + ((size_t)(b * N_ + n)) * 4;
    const float* b2 = tbxy + ((size_t)(b * M_ + col)) * 4;
    float x1 = b1[0], y1 = b1[1], x2 = b1[2], y2 = b1[3];
    float xg1 = b2[0], yg1 = b2[1], xg2 = b2[2], yg2 = b2[3];
    float xi1 = fmaxf(x1, xg1), yi1 = fmaxf(y1, yg1);
    float xi2 = fminf(x2, xg2), yi2 = fminf(y2, yg2);
    float inter = fmaxf(xi2 - xi1, 0.f) * fmaxf(yi2 - yi1, 0.f);
    float uni = (x2 - x1) * (y2 - y1) + (xg2 - xg1) * (yg2 - yg1) - inter;
    float iou_plain = inter / uni;                 // first return (no eps)
    float iou_e = inter / (uni + EPS_);
    float xc1 = fminf(x1, xg1), yc1 = fminf(y1, yg1);
    float xc2 = fmaxf(x2, xg2), yc2 = fmaxf(y2, yg2);
    float dx = xc2 - xc1, dy = yc2 - yc1;
    float diag = dx * dx + dy * dy + EPS_;
    float cdx = (x1 + x2 - xg1 - xg2) * 0.5f;
    float cdy = (y1 + y2 - yg1 - yg2) * 0.5f;
    float diou = 1.f - iou_e + (cdx * cdx + cdy * cdy) / diag;

    int idx = b * N_ + n;
    float bce = -(t_m * lnparr[idx] + (1.f - t_m) * ln1arr[idx]);

    const float* bp = bbox_pred + (size_t)idx * 4;
    const float* bt = bbox_target + ((size_t)(b * M_ + col)) * 4;
    float l1 = fabsf(bp[0] - bt[0]) + fabsf(bp[1] - bt[1]) +
               fabsf(bp[2] - bt[2]) + fabsf(bp[3] - bt[3]);

    __shared__ float acc[5][256];
    acc[0][n] = w;
    acc[1][n] = diou * w;
    acc[2][n] = iou_plain * w;
    acc[3][n] = bce;
    acc[4][n] = l1 * w;
    __syncthreads();
    for (int s = 128; s > 0; s >>= 1) {
        if (n < s)
            for (int q = 0; q < 5; ++q) acc[q][n] += acc[q][n + s];
        __syncthreads();
    }
    if (n == 0)
        for (int q = 0; q < 5; ++q) partials[b * 8 + q] = acc[q][0];
}

__global__ void finalize_kernel(const float* __restrict__ partials,
                                float* __restrict__ out) {
    __shared__ float acc[5][64];
    int t = threadIdx.x;
    for (int q = 0; q < 5; ++q) acc[q][t] = partials[t * 8 + q];
    __syncthreads();
    for (int s = 32; s > 0; s >>= 1) {
        if (t < s)
            for (int q = 0; q < 5; ++q) acc[q][t] += acc[q][t + s];
        __syncthreads();
    }
    if (t == 0) {
        float wsum = acc[0][0], sdiou = acc[1][0], siou = acc[2][0];
        float sbce = acc[3][0], sl1 = acc[4][0];
        float total = sdiou / wsum + sbce * (1.f / (float)(B_ * N_)) + sl1 / (wsum * 4.f);
        out[0] = total;
        out[1] = siou / wsum;
    }
}

extern "C" void kernel_launch(void* const* d_in, const int* in_sizes, int n_in,
                              void* d_out, int out_size, void* d_ws, size_t ws_size,
                              hipStream_t stream) {
    const float* labels_pred   = (const float*)d_in[0];  // [B,N,1]
    const float* bbox_pred     = (const float*)d_in[1];  // [B,N,4]
    const float* labels_target = (const float*)d_in[2];  // [B,M] float {0,1}
    const float* bbox_target   = (const float*)d_in[3];  // [B,M,4]

    float* ws = (float*)d_ws;
    float* pbxy     = ws;                       // B*N*4   = 65536
    float* tbxy     = pbxy + B_ * N_ * 4;       // B*M*4   = 65536
    float* dT       = tbxy + B_ * M_ * 4;       // N*B     = 16384
    float* lnparr   = dT + N_ * B_;             // B*N
    float* ln1arr   = lnparr + B_ * N_;         // B*N
    float* s1       = ln1arr + B_ * N_;         // N
    float* Lmat     = s1 + N_;                  // N*M     = 65536
    float* costbase = Lmat + N_ * M_;           // N*M     = 65536
    int*   cols     = (int*)(costbase + N_ * M_);   // B*N ints
    float* partials = (float*)(cols + B_ * N_);     // B*8

    prep_kernel<<<(B_ * N_ + 255) / 256, 256, 0, stream>>>(
        labels_pred, bbox_pred, bbox_target, pbxy, tbxy, dT, lnparr, ln1arr);
    s1_kernel<<<1, 256, 0, stream>>>(ln1arr, s1);
    label_gemm_kernel<<<(N_ / 16) * (M_ / 16), 32, 0, stream>>>(dT, labels_target, Lmat);
    costbase_kernel<<<(N_ / 16) * (M_ / 16), 256, 0, stream>>>(
        bbox_pred, bbox_target, Lmat, s1, costbase);
    hungarian_kernel<<<B_, 256, 0, stream>>>(pbxy, tbxy, costbase, cols);
    matched_kernel<<<B_, 256, 0, stream>>>(pbxy, tbxy, bbox_pred, bbox_target,
                                           labels_target, lnparr, ln1arr, cols, partials);
    finalize_kernel<<<1, 64, 0, stream>>>(partials, (float*)d_out);
}